// SparseGCNLayer_36764920054542
// MI455X (gfx1250) — compile-verified
//
#include <hip/hip_runtime.h>

typedef __attribute__((ext_vector_type(2))) float v2f;
typedef __attribute__((ext_vector_type(8))) float v8f;

#define N_NODES 100000
#define N_EDGES 3200000
#define D 128
#define M_TILES (N_NODES / 16)   // 6250 exactly
#define TI 8                     // i-tiles per wave in GEMM

// ---------------- Kernel 1: zero the propagated buffer ----------------
__global__ __launch_bounds__(256) void gcn_zero_kernel(float4* __restrict__ p, int n4) {
    int i = blockIdx.x * blockDim.x + threadIdx.x;
    if (i < n4) p[i] = make_float4(0.f, 0.f, 0.f, 0.f);
}

// ---------------- Kernel 2: SpMM scatter, one wave32 per edge ----------------
// prop[row] += val * x[col]   (row of 128 floats; 4 floats per lane)
__global__ __launch_bounds__(256) void gcn_scatter_kernel(
    const float* __restrict__ x,
    const long long* __restrict__ erow,
    const long long* __restrict__ ecol,
    const float* __restrict__ eval_,
    float* __restrict__ prop) {
    long long edge = (long long)((blockIdx.x * blockDim.x + threadIdx.x) >> 5);
    int lane = threadIdx.x & 31;
    if (edge >= N_EDGES) return;

    long long r = erow[edge];
    long long c = ecol[edge];
    float v = eval_[edge];

    // coalesced 512B gather of row x[c]
    const float4* xr = (const float4*)(x + c * (long long)D);
    float4 m = xr[lane];

    float* dst = prop + r * (long long)D + lane * 4;
    atomicAdd(dst + 0, v * m.x);
    atomicAdd(dst + 1, v * m.y);
    atomicAdd(dst + 2, v * m.z);
    atomicAdd(dst + 3, v * m.w);
}

// ---------------- Kernel 3: out = P @ W^T + b via V_WMMA_F32_16X16X4_F32 ----
// out[i,j] = sum_k P[i,k] * W[j,k] + b[j]
// Each wave: one 16-wide j-tile, B-fragments of W held in registers,
// reused across TI consecutive 16-row i-tiles.
__global__ __launch_bounds__(256) void gcn_gemm_wmma_kernel(
    const float* __restrict__ P,     // [N_NODES, D]
    const float* __restrict__ Wm,    // [D, D]
    const float* __restrict__ bias,  // [D]
    float* __restrict__ out) {       // [N_NODES, D]
    const int lane  = threadIdx.x & 31;
    const int gw    = blockIdx.x * 8 + (threadIdx.x >> 5);
    const int jt    = gw & 7;        // 8 j-tiles (D=128)
    const int strip = gw >> 3;       // strip of TI i-tiles
    const int n  = lane & 15;        // N index (and A's M index)
    const int kb = (lane >> 4) * 2;  // K sub-offset within a k-step
    const int j0 = jt * 16;

    // Preload B fragments: B[kk] covers K = 4*kk .. 4*kk+3 of W^T
    // B[kk][kb+t][n] = W[j0+n][4*kk+kb+t]  -> contiguous float2 per lane
    v2f bfrag[32];
#pragma unroll
    for (int kk = 0; kk < 32; ++kk) {
        bfrag[kk] = *(const v2f*)(Wm + (j0 + n) * D + kk * 4 + kb);
    }
    const float bv = bias[j0 + n];

    for (int t = 0; t < TI; ++t) {
        int it = strip * TI + t;
        if (it >= M_TILES) return;
        int i0 = it * 16;

        v8f acc = {};
        const float* prow = P + (long long)(i0 + n) * D + kb;  // A fragment base
#pragma unroll
        for (int kk = 0; kk < 32; ++kk) {
            v2f afrag = *(const v2f*)(prow + kk * 4);
            acc = __builtin_amdgcn_wmma_f32_16x16x4_f32(
                /*neg_a=*/false, afrag, /*neg_b=*/false, bfrag[kk],
                /*c_mod=*/(short)0, acc, /*reuse_a=*/false, /*reuse_b=*/false);
        }

        // C/D layout: lane l, vgpr v -> row i0 + v + 8*(l>>4), col j0 + (l&15)
        int rbase = i0 + (lane >> 4) * 8;
#pragma unroll
        for (int v = 0; v < 8; ++v) {
            out[(long long)(rbase + v) * D + j0 + n] = acc[v] + bv;
        }
    }
}

extern "C" void kernel_launch(void* const* d_in, const int* in_sizes, int n_in,
                              void* d_out, int out_size, void* d_ws, size_t ws_size,
                              hipStream_t stream) {
    const float*     x    = (const float*)d_in[0];
    const long long* erow = (const long long*)d_in[1];
    const long long* ecol = (const long long*)d_in[2];
    const float*     ev   = (const float*)d_in[3];
    const float*     Wm   = (const float*)d_in[4];
    const float*     b    = (const float*)d_in[5];
    float* out  = (float*)d_out;
    float* prop = (float*)d_ws;   // N_NODES * D floats = 51.2 MB scratch

    // 1) zero propagated
    int n4 = N_NODES * D / 4;
    gcn_zero_kernel<<<(n4 + 255) / 256, 256, 0, stream>>>((float4*)prop, n4);

    // 2) edge scatter: one wave per edge -> E*32 threads
    long long nthreads = (long long)N_EDGES * 32;
    int nblocks = (int)((nthreads + 255) / 256);
    gcn_scatter_kernel<<<nblocks, 256, 0, stream>>>(x, erow, ecol, ev, prop);

    // 3) WMMA GEMM + bias: waves = 8 j-tiles * ceil(6250/TI) strips, 8 waves/block
    int strips = (M_TILES + TI - 1) / TI;           // 782
    gcn_gemm_wmma_kernel<<<strips, 256, 0, stream>>>(prop, Wm, b, out);
}